// SAGE_20383914787326
// MI455X (gfx1250) — compile-verified
//
#include <hip/hip_runtime.h>

#define N_NODES 100000
#define N_EDGES 1600000
#define D 128
#define KK 256            // concatenated K: 128 self + 128 neigh
#define LDS_STRIDE 258    // 256 + 2 pad -> lane rows land on distinct banks

typedef float v2f __attribute__((ext_vector_type(2)));
typedef float v8f __attribute__((ext_vector_type(8)));

__global__ void zero_f32(float* __restrict__ p, int n) {
    int i = blockIdx.x * blockDim.x + threadIdx.x;
    int stride = gridDim.x * blockDim.x;
    for (; i < n; i += stride) p[i] = 0.0f;
}

__global__ void degree_kernel(const int* __restrict__ dst, float* __restrict__ deg) {
    int e = blockIdx.x * blockDim.x + threadIdx.x;
    if (e < N_EDGES) atomicAdd(&deg[dst[e]], 1.0f);
}

__global__ void invdeg_kernel(float* __restrict__ deg) {
    int i = blockIdx.x * blockDim.x + threadIdx.x;
    if (i < N_NODES) deg[i] = 1.0f / fmaxf(deg[i], 1.0f);
}

// One wave32 per edge; lane l moves features [4l, 4l+3] (32*4 = 128).
__global__ void scatter_kernel(const float* __restrict__ h, const int* __restrict__ src,
                               const int* __restrict__ dst, float* __restrict__ agg) {
    int wave = threadIdx.x >> 5;
    int lane = threadIdx.x & 31;
    int e = blockIdx.x * 8 + wave;
    if (e >= N_EDGES) return;
    int s = src[e];
    int d = dst[e];
    float4 v = ((const float4*)(h + (size_t)s * D))[lane];
    float* a = agg + (size_t)d * D + lane * 4;
    atomicAdd(a + 0, v.x);
    atomicAdd(a + 1, v.y);
    atomicAdd(a + 2, v.z);
    atomicAdd(a + 3, v.w);
}

// out[16 rows x n_out] = relu([X | agg*inv_deg] @ [Ws;Wn]^T + b)
// One WG per 16-row tile; wave w handles col tile w. Safe for out == X
// (all X reads staged to LDS before any store; rows private per WG).
__launch_bounds__(256)
__global__ void sage_gemm(const float* __restrict__ X, const float* __restrict__ agg,
                          const float* __restrict__ inv_deg,
                          const float* __restrict__ Ws, const float* __restrict__ Wn,
                          const float* __restrict__ bias,
                          float* __restrict__ out, int n_out, int n_col_tiles) {
    __shared__ float lds[16 * LDS_STRIDE];
    const int row0 = blockIdx.x * 16;
    const int tid = threadIdx.x;

    // Stage A panel: 16 rows x 256 "k" (first 128 = self, last 128 = agg * inv_deg)
    for (int i = tid; i < 16 * KK; i += 256) {
        int m = i >> 8;        // i / 256
        int k = i & 255;
        float v;
        if (k < D) v = X[(size_t)(row0 + m) * D + k];
        else       v = agg[(size_t)(row0 + m) * D + (k - D)] * inv_deg[row0 + m];
        lds[m * LDS_STRIDE + k] = v;
    }
    __syncthreads();

    const int wave = tid >> 5;
    if (wave >= n_col_tiles) return;     // wave-uniform exit (layer 2 uses 3 tiles)
    const int lane = tid & 31;
    const int mloc = lane & 15;          // A: row within tile / B: col within tile
    const int khi  = lane >> 4;          // selects K pair per ISA 16x4 f32 layout
    const int col  = wave * 16 + mloc;
    const bool colvalid = col < n_out;
    const float* wsrow = Ws + (size_t)min(col, n_out - 1) * D;
    const float* wnrow = Wn + (size_t)min(col, n_out - 1) * D;

    v8f acc = {0.f, 0.f, 0.f, 0.f, 0.f, 0.f, 0.f, 0.f};
    #pragma unroll 8
    for (int kk = 0; kk < KK; kk += 4) {
        // A frag: lane(m = lane&15, k = kk + 2*(lane>>4) + v)
        v2f a = *(const v2f*)&lds[mloc * LDS_STRIDE + kk + 2 * khi];
        // B frag: B[k][n] = W[col][k], same lane/k striping as A
        v2f b;
        if (kk < D) b = *(const v2f*)&wsrow[kk + 2 * khi];
        else        b = *(const v2f*)&wnrow[kk - D + 2 * khi];
        if (!colvalid) { b.x = 0.0f; b.y = 0.0f; }
        acc = __builtin_amdgcn_wmma_f32_16x16x4_f32(
            /*neg_a=*/false, a, /*neg_b=*/false, b,
            /*c_mod=*/(short)0, acc, /*reuse_a=*/false, /*reuse_b=*/false);
    }

    if (colvalid) {
        float bv = bias[col];
        #pragma unroll
        for (int r = 0; r < 8; ++r) {
            int row = row0 + r + 8 * khi;         // D layout: vgpr r -> M = r, r+8
            float v = acc[r] + bv;
            v = v > 0.0f ? v : 0.0f;
            out[(size_t)row * n_out + col] = v;
        }
    }
}

extern "C" void kernel_launch(void* const* d_in, const int* in_sizes, int n_in,
                              void* d_out, int out_size, void* d_ws, size_t ws_size,
                              hipStream_t stream) {
    (void)in_sizes; (void)n_in; (void)out_size; (void)ws_size;
    const float* x   = (const float*)d_in[0];
    /* d_in[1] = edge_weight: unused by reference */
    const int*   src = (const int*)d_in[2];
    const int*   dst = (const int*)d_in[3];
    const float* Ws0 = (const float*)d_in[4];
    const float* Wn0 = (const float*)d_in[5];
    const float* b0  = (const float*)d_in[6];
    const float* Ws1 = (const float*)d_in[7];
    const float* Wn1 = (const float*)d_in[8];
    const float* b1  = (const float*)d_in[9];
    const float* Ws2 = (const float*)d_in[10];
    const float* Wn2 = (const float*)d_in[11];
    const float* b2  = (const float*)d_in[12];
    float* out = (float*)d_out;

    // Workspace layout (all 16B-aligned): inv_deg | agg | h  ~= 98 MB
    float* inv_deg = (float*)d_ws;
    float* agg = inv_deg + N_NODES;
    float* h   = agg + (size_t)N_NODES * D;

    const int row_tiles = N_NODES / 16;       // 6250 exact
    const int edge_blks = N_EDGES / 8;        // 200000 exact

    // degrees -> inv_deg (in place)
    zero_f32<<<256, 256, 0, stream>>>(inv_deg, N_NODES);
    degree_kernel<<<(N_EDGES + 255) / 256, 256, 0, stream>>>(dst, inv_deg);
    invdeg_kernel<<<(N_NODES + 255) / 256, 256, 0, stream>>>(inv_deg);

    // layer 0: x -> h
    zero_f32<<<2048, 256, 0, stream>>>(agg, N_NODES * D);
    scatter_kernel<<<edge_blks, 256, 0, stream>>>(x, src, dst, agg);
    sage_gemm<<<row_tiles, 256, 0, stream>>>(x, agg, inv_deg, Ws0, Wn0, b0, h, D, 8);

    // layer 1: h -> h (in place, safe: per-WG rows staged in LDS before store)
    zero_f32<<<2048, 256, 0, stream>>>(agg, N_NODES * D);
    scatter_kernel<<<edge_blks, 256, 0, stream>>>(h, src, dst, agg);
    sage_gemm<<<row_tiles, 256, 0, stream>>>(h, agg, inv_deg, Ws1, Wn1, b1, h, D, 8);

    // layer 2: h -> out (40 classes, 3 col tiles with masking)
    zero_f32<<<2048, 256, 0, stream>>>(agg, N_NODES * D);
    scatter_kernel<<<edge_blks, 256, 0, stream>>>(h, src, dst, agg);
    sage_gemm<<<row_tiles, 256, 0, stream>>>(h, agg, inv_deg, Ws2, Wn2, b2, out, 40, 3);
}